// Slicing_CUDA_45217415692504
// MI455X (gfx1250) — compile-verified
//
#include <hip/hip_runtime.h>

// Bilateral-grid slicing (trilinear) for MI455X / gfx1250.
// grid:  [B, c, D=8, Gh=16, Gw=16] f32   (8 KB per (b,c) slice)
// guide: [B, c, H=1024, W=1024]  f32
// out:   [B, c, H, W] f32
//
// Memory-bound gather op: stage the 8 KB grid slice in LDS via the CDNA5
// async-to-LDS path, stream guide/out with non-temporal B128 accesses.

#define BLOCK   256      // 8 wave32 waves per block
#define PPT     8        // pixels per thread (2 x float4)
#define W_IMG   1024
#define GW      16
#define GH      16
#define GD      8
#define SLICE   (GD * GH * GW)   // 2048 floats per (b,c) slice

typedef float v4f __attribute__((ext_vector_type(4)));
typedef int   v4i __attribute__((vector_size(16)));   // type the async builtin wants

__global__ __launch_bounds__(BLOCK) void slice_trilinear_kernel(
    const float* __restrict__ grid,
    const float* __restrict__ guide,
    float* __restrict__ out,
    int HW)
{
    __shared__ float sg[SLICE];

    const int tid = threadIdx.x;
    const int bc  = blockIdx.y;
    const float* gsl = grid + (size_t)bc * SLICE;

    // ---- Stage the 8 KB grid slice into LDS -------------------------------
    // 2048 floats = 512 x B128; 256 threads x 2 async B128 copies.
#if __has_builtin(__builtin_amdgcn_global_load_async_to_lds_b128)
    {
        float* gsrc = const_cast<float*>(gsl);
        __builtin_amdgcn_global_load_async_to_lds_b128(
            (v4i*)(gsrc + tid * 4),        (v4i*)(sg + tid * 4),        0, 0);
        __builtin_amdgcn_global_load_async_to_lds_b128(
            (v4i*)(gsrc + 1024 + tid * 4), (v4i*)(sg + 1024 + tid * 4), 0, 0);
        asm volatile("s_wait_asynccnt 0" ::: "memory");
    }
#else
    {
        const v4f* src = (const v4f*)gsl;
        v4f*       dst = (v4f*)sg;
        dst[tid]         = src[tid];
        dst[tid + BLOCK] = src[tid + BLOCK];
    }
#endif
    __syncthreads();

    // ---- Per-pixel trilinear slice ---------------------------------------
    const size_t base = (size_t)bc * (size_t)HW;
    const int p0 = (blockIdx.x * BLOCK + tid) * PPT;

    // align_corners mapping: ix = x * (Gw-1)/W  (same for y; square image)
    const float sxy = (float)(GW - 1) / (float)W_IMG;

    #pragma unroll
    for (int v = 0; v < PPT / 4; ++v) {
        const int p = p0 + v * 4;
        if (p >= HW) break;

        // streaming read: non-temporal, keep L2 for the grid
        const v4f g4 = __builtin_nontemporal_load((const v4f*)(guide + base + p));
        const float gv[4] = {g4.x, g4.y, g4.z, g4.w};
        float rv[4];

        #pragma unroll
        for (int k = 0; k < 4; ++k) {
            const int pp = p + k;
            const int x = pp & (W_IMG - 1);
            const int y = pp >> 10;                 // W = 1024

            const float fx = (float)x * sxy;        // in [0, 15)
            const float fy = (float)y * sxy;        // in [0, 15)
            float fz = (gv[k] + 1.0f) * (0.5f * (float)(GD - 1));
            fz = fminf(fmaxf(fz, 0.0f), (float)(GD - 1));

            const int ix0 = (int)fx; const float wx = fx - (float)ix0;  // ix0 <= 14
            const int iy0 = (int)fy; const float wy = fy - (float)iy0;  // iy0 <= 14
            const int iz0 = (int)fz; const float wz = fz - (float)iz0;
            int iz1 = iz0 + 1; if (iz1 > GD - 1) iz1 = GD - 1;

            // ix1 = ix0+1, iy1 = iy0+1 are always in range (fx,fy < 15)
            const int b00 = iz0 * (GH * GW) + iy0 * GW + ix0;
            const int b01 = b00 + GW;
            const int b10 = iz1 * (GH * GW) + iy0 * GW + ix0;
            const int b11 = b10 + GW;

            const float c000 = sg[b00],     c001 = sg[b00 + 1];
            const float c010 = sg[b01],     c011 = sg[b01 + 1];
            const float c100 = sg[b10],     c101 = sg[b10 + 1];
            const float c110 = sg[b11],     c111 = sg[b11 + 1];

            const float c00 = fmaf(wx, c001 - c000, c000);
            const float c01 = fmaf(wx, c011 - c010, c010);
            const float c10 = fmaf(wx, c101 - c100, c100);
            const float c11 = fmaf(wx, c111 - c110, c110);
            const float c0  = fmaf(wy, c01 - c00, c00);
            const float c1  = fmaf(wy, c11 - c10, c10);
            rv[k] = fmaf(wz, c1 - c0, c0);
        }

        v4f o4;
        o4.x = rv[0]; o4.y = rv[1]; o4.z = rv[2]; o4.w = rv[3];
        __builtin_nontemporal_store(o4, (v4f*)(out + base + p));
    }
}

extern "C" void kernel_launch(void* const* d_in, const int* in_sizes, int n_in,
                              void* d_out, int out_size, void* d_ws, size_t ws_size,
                              hipStream_t stream) {
    const float* grid  = (const float*)d_in[0];   // [B, c, 8, 16, 16]
    const float* guide = (const float*)d_in[1];   // [B, c, 1024, 1024]
    float*       out   = (float*)d_out;

    const int bc = in_sizes[0] / SLICE;           // B*c = 12
    const int HW = in_sizes[1] / bc;              // 1024*1024

    const int pix_per_block = BLOCK * PPT;        // 2048
    dim3 g((HW + pix_per_block - 1) / pix_per_block, bc);
    slice_trilinear_kernel<<<g, dim3(BLOCK), 0, stream>>>(grid, guide, out, HW);
}